// OUProcess_10307921510654
// MI455X (gfx1250) — compile-verified
//
#include <hip/hip_runtime.h>
#include <cstdint>

// OU backward substitution: x[i] = (z[i] - sup[i+1]*x[i+1]) / diag[i], per row.
// rows = 32*256 = 8192, L = 8192, f32. Bandwidth-bound (0.25 flop/byte) on the
// 23.3 TB/s HBM roofline -> optimize the data path: async global<->LDS tile
// staging (CDNA5 ASYNCcnt), triple-buffered prefetch, segmented affine scan for
// 8x wave-level parallelism. WMMA is irrelevant at this arithmetic intensity.

namespace {
constexpr int ROWS  = 32 * 256;   // 8192 independent rows
constexpr int LL    = 8192;       // recurrence length
constexpr int S     = 8;          // segments per row
constexpr int SEG   = LL / S;     // 1024
constexpr int TT    = 64;         // tile columns
constexpr int NT    = SEG / TT;   // 16 tiles per segment
constexpr int PITCH = 68;         // LDS row pitch in floats: 272B = 16B-aligned rows
constexpr int RGRP  = ROWS / 32;  // 256 row groups (32 rows per wave)
static_assert(SEG % TT == 0, "");
}

__device__ __forceinline__ uint32_t lds_off(const void* p) {
  // Low 32 bits of a generic pointer into LDS are the DS-relative byte offset
  // (CDNA5 ISA 10.2).
  return (uint32_t)(uintptr_t)p;
}

// One b128 async op moves 2 tile rows: lanes 0-15 -> row 2k, lanes 16-31 -> row
// 2k+1, 16 contiguous bytes per lane. 16 ops per 32x64 f32 tile. GVS addressing
// (SGPR base + 32-bit VGPR offset; max offset 256MB < 2^31).
__device__ __forceinline__ void async_load_tile(const float* __restrict__ gp,
                                                uint32_t ldsbase, int row0,
                                                int col, int lane) {
  const int sub = lane >> 4;
  const int c4  = (lane & 15) << 2;
  uint32_t gbase = ((uint32_t)(row0 + sub) * (uint32_t)LL + (uint32_t)(col + c4)) * 4u;
  uint32_t lbase = ldsbase + (uint32_t)((sub * PITCH + c4) * 4);
#pragma unroll
  for (int k = 0; k < 16; ++k) {
    uint32_t goff = gbase + (uint32_t)k * (uint32_t)(2 * LL * 4);
    uint32_t loff = lbase + (uint32_t)k * (uint32_t)(2 * PITCH * 4);
    asm volatile("global_load_async_to_lds_b128 %0, %1, %2"
                 :: "v"(loff), "v"(goff), "s"(gp) : "memory");
  }
}

__device__ __forceinline__ void async_store_tile(float* __restrict__ gp,
                                                 uint32_t ldsbase, int row0,
                                                 int col, int lane) {
  const int sub = lane >> 4;
  const int c4  = (lane & 15) << 2;
  uint32_t gbase = ((uint32_t)(row0 + sub) * (uint32_t)LL + (uint32_t)(col + c4)) * 4u;
  uint32_t lbase = ldsbase + (uint32_t)((sub * PITCH + c4) * 4);
#pragma unroll
  for (int k = 0; k < 16; ++k) {
    uint32_t goff = gbase + (uint32_t)k * (uint32_t)(2 * LL * 4);
    uint32_t loff = lbase + (uint32_t)k * (uint32_t)(2 * PITCH * 4);
    asm volatile("global_store_async_from_lds_b128 %0, %1, %2"
                 :: "v"(goff), "v"(loff), "s"(gp) : "memory");
  }
}

// Wait threshold 16*min(t,2). With triple-buffered loads AND stores, and
// per-type in-order completion: at iteration t, pending loads are from tiles
// {t, t-1, t-2}, pending stores from {t+1, t+2, t+3} (induction). If loads(t)
// were incomplete, >=33 load-ops remain; if stores(t+3) were incomplete, >=33
// store-ops remain. Either contradicts ASYNCcnt <= 32.
__device__ __forceinline__ void wait_async(int t) {
  if (t >= 2)      asm volatile("s_wait_asynccnt 32" ::: "memory");
  else if (t == 1) asm volatile("s_wait_asynccnt 16" ::: "memory");
  else             asm volatile("s_wait_asynccnt 0"  ::: "memory");
}

// Pass 1: per (32-row group, segment) wave, compose the segment's affine map
// x_segstart = A * x_segend + B by reverse-iterating (A,B) <- (c*A, c*B + g),
// c_i = -e_i*rd_i, g_i = z_i*rd_i. z staged via triple-buffered async-LDS tiles;
// segment coefficients (1/diag, sup shifted) preloaded to LDS once, overlapping
// the prologue DMA.
__global__ __launch_bounds__(32) void ou_pass1(const float* __restrict__ z,
                                               const float* __restrict__ dg,
                                               const float* __restrict__ sp,
                                               float* __restrict__ Aout,
                                               float* __restrict__ Bout) {
  __shared__ float zb[3][32][PITCH];
  __shared__ float crd[SEG];
  __shared__ float ce[SEG];
  const int lane = threadIdx.x;
  const int rg   = blockIdx.x & (RGRP - 1);
  const int seg  = blockIdx.x >> 8;
  const int row0 = rg * 32;
  const int seg0 = seg * SEG;
  const uint32_t zbo[3] = { lds_off(&zb[0][0][0]), lds_off(&zb[1][0][0]),
                            lds_off(&zb[2][0][0]) };

  async_load_tile(z, zbo[(NT - 1) % 3], row0, seg0 + (NT - 1) * TT, lane);
  async_load_tile(z, zbo[(NT - 2) % 3], row0, seg0 + (NT - 2) * TT, lane);

  // Coefficient preload (coalesced, shared by all 32 rows), hides DMA latency.
  for (int j = lane; j < SEG; j += 32) {
    const int i = seg0 + j;
    crd[j] = 1.0f / dg[i];
    ce[j]  = (i + 1 < LL) ? sp[i + 1] : 0.0f;
  }

  float A = 1.0f, Bacc = 0.0f;
  for (int t = NT - 1; t >= 0; --t) {
    if (t >= 2) {
      asm volatile("s_wait_dscnt 0" ::: "memory");   // buffer fully consumed
      async_load_tile(z, zbo[(t - 2) % 3], row0, seg0 + (t - 2) * TT, lane);
    }
    wait_async(t);
    const float* zr  = &zb[t % 3][lane][0];
    const float* rdp = &crd[t * TT];
    const float* ep  = &ce[t * TT];
#pragma unroll 8
    for (int i = TT - 1; i >= 0; --i) {
      float rd = rdp[i];
      float e  = ep[i];
      float zv = zr[i];
      float c  = -(e * rd);
      Bacc = fmaf(c, Bacc, zv * rd);
      A    = c * A;
    }
  }
  const int row = row0 + lane;
  Aout[seg * ROWS + row] = A;
  Bout[seg * ROWS + row] = Bacc;
}

// Pass 2: per row, sequentially combine S=8 segment maps right-to-left to get
// each segment's incoming boundary value x[(s+1)*SEG] (0 for the last segment:
// e[L-1]=0 makes A[S-1]'s incoming irrelevant). Layout [seg][row] -> coalesced.
__global__ __launch_bounds__(256) void ou_pass2(const float* __restrict__ A,
                                                const float* __restrict__ B,
                                                float* __restrict__ Xin) {
  const int row = blockIdx.x * blockDim.x + threadIdx.x;
  float inc = 0.0f;
#pragma unroll
  for (int s = S - 1; s >= 0; --s) {
    const int idx = s * ROWS + row;
    Xin[idx] = inc;
    inc = fmaf(A[idx], inc, B[idx]);
  }
}

// Pass 3: re-solve each segment with its known incoming x; z tiles async-loaded
// (triple buffer), x tiles written back with global_store_async_from_lds_b128
// (triple buffer, see wait_async proof). s_endpgm's implicit wait-idle drains
// the trailing stores.
__global__ __launch_bounds__(32) void ou_pass3(const float* __restrict__ z,
                                               const float* __restrict__ dg,
                                               const float* __restrict__ sp,
                                               const float* __restrict__ Xin,
                                               float* __restrict__ out) {
  __shared__ float zb[3][32][PITCH];
  __shared__ float xb[3][32][PITCH];
  __shared__ float crd[SEG];
  __shared__ float ce[SEG];
  const int lane = threadIdx.x;
  const int rg   = blockIdx.x & (RGRP - 1);
  const int seg  = blockIdx.x >> 8;
  const int row0 = rg * 32;
  const int seg0 = seg * SEG;
  const uint32_t zbo[3] = { lds_off(&zb[0][0][0]), lds_off(&zb[1][0][0]),
                            lds_off(&zb[2][0][0]) };
  const uint32_t xbo[3] = { lds_off(&xb[0][0][0]), lds_off(&xb[1][0][0]),
                            lds_off(&xb[2][0][0]) };

  async_load_tile(z, zbo[(NT - 1) % 3], row0, seg0 + (NT - 1) * TT, lane);
  async_load_tile(z, zbo[(NT - 2) % 3], row0, seg0 + (NT - 2) * TT, lane);

  float x = Xin[seg * ROWS + row0 + lane];   // incoming x[(seg+1)*SEG]

  for (int j = lane; j < SEG; j += 32) {
    const int i = seg0 + j;
    crd[j] = 1.0f / dg[i];
    ce[j]  = (i + 1 < LL) ? sp[i + 1] : 0.0f;
  }

  for (int t = NT - 1; t >= 0; --t) {
    const int col = seg0 + t * TT;
    if (t >= 2) {
      asm volatile("s_wait_dscnt 0" ::: "memory");
      async_load_tile(z, zbo[(t - 2) % 3], row0, col - 2 * TT, lane);
    }
    wait_async(t);   // tile t's loads done AND tile t+3's store buffer free
    const float* zr  = &zb[t % 3][lane][0];
    float*       xr  = &xb[t % 3][lane][0];
    const float* rdp = &crd[t * TT];
    const float* ep  = &ce[t * TT];
#pragma unroll 8
    for (int i = TT - 1; i >= 0; --i) {
      x = (zr[i] - ep[i] * x) * rdp[i];
      xr[i] = x;
    }
    asm volatile("s_wait_dscnt 0" ::: "memory");       // xb writes visible to DMA
    async_store_tile(out, xbo[t % 3], row0, col, lane);
  }
}

extern "C" void kernel_launch(void* const* d_in, const int* in_sizes, int n_in,
                              void* d_out, int out_size, void* d_ws, size_t ws_size,
                              hipStream_t stream) {
  const float* z  = (const float*)d_in[0];   // (32, 256, 8192) f32
  const float* dg = (const float*)d_in[1];   // (8192,) f32
  const float* sp = (const float*)d_in[2];   // (8192,) f32
  float* out = (float*)d_out;

  // Workspace: A, B, Xin each ROWS*S floats = 3 * 256 KiB = 768 KiB.
  float* A   = (float*)d_ws;
  float* B   = A + (size_t)ROWS * S;
  float* Xin = B + (size_t)ROWS * S;

  const int nblk = RGRP * S;   // 2048 one-wave workgroups
  ou_pass1<<<nblk, 32, 0, stream>>>(z, dg, sp, A, B);
  ou_pass2<<<ROWS / 256, 256, 0, stream>>>(A, B, Xin);
  ou_pass3<<<nblk, 32, 0, stream>>>(z, dg, sp, Xin, out);
}